// CombinedModel_74655121539887
// MI455X (gfx1250) — compile-verified
//
#include <hip/hip_runtime.h>

// ---------------------------------------------------------------------------
// YOLO decode, MI455X (gfx1250). Memory-bound streaming transform:
//   274 MB total traffic / 23.3 TB/s  =>  ~11.8 us floor.
// Strategy:
//   - stage input via CDNA5 async global->LDS B128 copies (ASYNCcnt path)
//   - s_wait_asynccnt 0 + workgroup barrier, compute from LDS (ds_load)
//   - fully branchless channel decode (v_cndmask selects, no exec-mask
//     scaffolding: every wave straddles box+sigmoid channels anyway)
//   - packed 128-bit non-temporal stores (write-once output, keep it out
//     of the 192 MB L2)
// ---------------------------------------------------------------------------

#define TPB          256
#define ANCH_PER_WG  48
#define CH           85
#define SMEM_FLOATS  (ANCH_PER_WG * CH)          // 4080 floats = 16320 B
#define B128_CHUNKS  (SMEM_FLOATS * 4 / 16)      // 1020 16-byte granules

typedef __attribute__((address_space(3))) float lds_f32;
typedef __attribute__((ext_vector_type(4)))  float v4f;

__global__ __launch_bounds__(TPB) void yolo_decode_map(
    const float* __restrict__ src,   // this map's fm, [B, Nm, 85] flat
    float* __restrict__ dst,         // full output,   [B, Ntot, 85] flat
    int anchorsPerBatch,             // Nm (multiple of 48)
    int anchorOffset,                // Om within output anchor dim
    int totalAnchorsOut,             // Ntot = 25200
    float wscale, float hscale)
{
    __shared__ alignas(16) float smem[SMEM_FLOATS];

    const int b     = blockIdx.y;
    const int chunk = blockIdx.x;
    const int tid   = threadIdx.x;
    __builtin_assume(tid < TPB);     // kills redundant predicate blocks

    const size_t srcBase =
        ((size_t)b * anchorsPerBatch + (size_t)chunk * ANCH_PER_WG) * CH;
    const size_t dstBase =
        ((size_t)b * totalAnchorsOut + (size_t)anchorOffset +
         (size_t)chunk * ANCH_PER_WG) * CH;

    const float* srcChunk = src + srcBase;        // 16-byte aligned by construction
    // LDS byte offset of smem (addrspace(3) ptrtoint == raw LDS offset).
    const unsigned ldsBase = (unsigned)(unsigned long long)(lds_f32*)smem;

    // ---- Stage 1: async global -> LDS, 128-bit granules --------------------
    #pragma unroll
    for (int i = 0; i < (B128_CHUNKS + TPB - 1) / TPB; ++i) {
        int idx = tid + i * TPB;
        if (idx < B128_CHUNKS) {                   // only live for last iter
            unsigned goff = (unsigned)idx * 16u;   // byte offset into chunk
            unsigned loff = ldsBase + goff;        // LDS byte address
            asm volatile("global_load_async_to_lds_b128 %0, %1, %2"
                         :: "v"(loff), "v"(goff), "s"(srcChunk)
                         : "memory");
        }
    }
    // Wait own async loads, then make LDS visible across all 8 waves.
    asm volatile("s_wait_asynccnt 0" ::: "memory");
    __syncthreads();

    // ---- Stage 2: branchless transform from LDS, 128-bit NT stores ---------
    float* dstChunk = dst + dstBase;               // 16-byte aligned
    #pragma unroll
    for (int i = 0; i < (B128_CHUNKS + TPB - 1) / TPB; ++i) {
        int q = tid + i * TPB;                     // float4 index in chunk
        if (q < B128_CHUNKS) {                     // only live for last iter
            v4f vv;
            #pragma unroll
            for (int j = 0; j < 4; ++j) {
                int e = 4 * q + j;                 // flat element in chunk
                int a = e / CH;                    // anchor within chunk
                int c = e - a * CH;                // channel
                const float* row = smem + a * CH;

                // sigmoid candidate (channels 4..84)
                float x   = row[c];
                float sig = 1.0f / (1.0f + __expf(-x));

                // box candidates (channels 0..3)
                float b0 = row[0], b1 = row[1], b2 = row[2], b3 = row[3];
                float x1 = (b0 - 0.5f * b2) * wscale;
                float y1 = (b1 - 0.5f * b3) * hscale;
                float x2 = (x1 + 0.5f * b2) * wscale;
                float y2 = (y1 + 0.5f * b3) * hscale;

                // branchless 4-way + final select (v_cndmask chain)
                float bx  = (c & 2) ? x2 : x1;
                float by  = (c & 2) ? y2 : y1;
                float box = (c & 1) ? by : bx;
                vv[j] = (c < 4) ? box : sig;
            }
            __builtin_nontemporal_store(vv, (v4f*)dstChunk + q);
        }
    }
}

extern "C" void kernel_launch(void* const* d_in, const int* in_sizes, int n_in,
                              void* d_out, int out_size, void* d_ws, size_t ws_size,
                              hipStream_t stream) {
    (void)in_sizes; (void)n_in; (void)out_size; (void)d_ws; (void)ws_size;

    const float* fm0 = (const float*)d_in[0];  // [16,3,80,80,85]
    const float* fm1 = (const float*)d_in[1];  // [16,3,40,40,85]
    const float* fm2 = (const float*)d_in[2];  // [16,3,20,20,85]
    float* out = (float*)d_out;                // [16,25200,85]

    const int B    = 16;
    const int N0   = 3 * 80 * 80;   // 19200 anchors/batch
    const int N1   = 3 * 40 * 40;   //  4800
    const int N2   = 3 * 20 * 20;   //  1200
    const int Ntot = N0 + N1 + N2;  // 25200

    dim3 blk(TPB);
    yolo_decode_map<<<dim3(N0 / ANCH_PER_WG, B), blk, 0, stream>>>(
        fm0, out, N0, 0,       Ntot, 80.0f, 80.0f);
    yolo_decode_map<<<dim3(N1 / ANCH_PER_WG, B), blk, 0, stream>>>(
        fm1, out, N1, N0,      Ntot, 40.0f, 40.0f);
    yolo_decode_map<<<dim3(N2 / ANCH_PER_WG, B), blk, 0, stream>>>(
        fm2, out, N2, N0 + N1, Ntot, 20.0f, 20.0f);
}